// PointPillarScatter_41257455846129
// MI455X (gfx1250) — compile-verified
//
#include <hip/hip_runtime.h>
#include <cstdint>
#include <cstddef>

// PointPillars BEV scatter, MI455X (gfx1250).
// Strategy: inverse-map gather. Scattered traffic is redirected onto the
// 192MB L2 (features table = 30.7MB, map = 6.9MB both L2-resident);
// the 439MB output is written exactly once with coalesced nontemporal
// float4 stores. ~500MB HBM traffic total -> ~21us at 23.3TB/s.

#define NXc   432
#define NYc   496
#define CBEV  64
#define PLANE (NXc * NYc)   // 214272

typedef __attribute__((ext_vector_type(4))) float f4;
typedef __attribute__((ext_vector_type(4))) int   i4;

// ---------------- K1: map[i] = -1 ----------------
__global__ void k_init_map(int* __restrict__ map, int n4) {
    int i = blockIdx.x * blockDim.x + threadIdx.x;
    if (i < n4) {
        i4 m = { -1, -1, -1, -1 };
        ((i4*)map)[i] = m;
    }
}

// ---------------- K2: map[gidx[p]] = p ----------------
__global__ void k_scatter_pid(const int* __restrict__ coords,
                              int* __restrict__ map, int P) {
    int p = blockIdx.x * blockDim.x + threadIdx.x;
    if (p < P) {
        int b = coords[4 * p + 0];
        int y = coords[4 * p + 2];
        int x = coords[4 * p + 3];
        map[(size_t)b * PLANE + (size_t)y * NXc + x] = p;
    }
}

// ---------------- K3: coalesced gather ----------------
// One block per (b, y, 16-wide x tile). 256 threads = 64 channels x 4 float4
// columns. Map segment (64B) staged into LDS once via the CDNA5 async
// global->LDS path and reused by all 64 channels.
__global__ void k_gather(const float* __restrict__ feat,
                         const int* __restrict__ map,
                         float* __restrict__ out) {
    __shared__ __align__(16) int smap[16];

    const unsigned TPR = NXc / 16;          // 27 tiles per row
    unsigned bid = blockIdx.x;
    unsigned t   = bid % TPR;
    unsigned y   = (bid / TPR) % NYc;
    unsigned b   = bid / (TPR * NYc);
    const unsigned x0  = t * 16;
    const unsigned tid = threadIdx.x;

    if (tid < 4) {
        const int* g = map + (size_t)b * PLANE + (size_t)y * NXc + x0 + tid * 4;
        uint64_t ga = (uint64_t)(uintptr_t)g;
        uint32_t lo = (uint32_t)(uintptr_t)(&smap[tid * 4]); // LDS byte offset
        asm volatile("global_load_async_to_lds_b128 %0, %1, off"
                     :: "v"(lo), "v"(ga) : "memory");
        asm volatile("s_wait_asynccnt 0" ::: "memory");
    }
    __syncthreads();

    const int c  = (int)(tid >> 2);        // channel 0..63
    const int xq = (int)(tid & 3) * 4;     // 0,4,8,12

    f4 v;
#pragma unroll
    for (int j = 0; j < 4; ++j) {
        int pid = smap[xq + j];
        v[j] = (pid >= 0) ? feat[(size_t)pid * CBEV + c] : 0.0f;
    }
    f4* dst = (f4*)(out + ((((size_t)b * CBEV + c) * NYc + y) * NXc + x0 + xq));
    __builtin_nontemporal_store(v, dst);   // stream past L2
}

// ---------------- Fallback path (ws too small): zero + direct scatter ----
__global__ void k_zero(float* __restrict__ out, int n4) {
    int i = blockIdx.x * blockDim.x + threadIdx.x;
    if (i < n4) {
        f4 z = { 0.f, 0.f, 0.f, 0.f };
        __builtin_nontemporal_store(z, (f4*)out + i);
    }
}

__global__ void k_scatter_direct(const float* __restrict__ feat,
                                 const int* __restrict__ coords,
                                 float* __restrict__ out, int PC) {
    int tidg = blockIdx.x * blockDim.x + threadIdx.x;
    if (tidg < PC) {
        int p = tidg >> 6;
        int c = tidg & 63;
        int b = coords[4 * p + 0];
        int y = coords[4 * p + 2];
        int x = coords[4 * p + 3];
        out[(((size_t)b * CBEV + c) * NYc + y) * NXc + x] = feat[tidg];
    }
}

extern "C" void kernel_launch(void* const* d_in, const int* in_sizes, int n_in,
                              void* d_out, int out_size, void* d_ws, size_t ws_size,
                              hipStream_t stream) {
    const float* feat   = (const float*)d_in[0];
    const int*   coords = (const int*)d_in[1];
    float*       out    = (float*)d_out;

    const int P  = in_sizes[0] / CBEV;             // 120000
    const int Bn = out_size / (CBEV * PLANE);      // 8
    const size_t mapBytes = (size_t)Bn * PLANE * sizeof(int);

    if (ws_size >= mapBytes) {
        int* map = (int*)d_ws;
        int n4 = (Bn * PLANE) / 4;
        k_init_map<<<(n4 + 255) / 256, 256, 0, stream>>>(map, n4);
        k_scatter_pid<<<(P + 255) / 256, 256, 0, stream>>>(coords, map, P);
        unsigned blocks = (unsigned)Bn * NYc * (NXc / 16);
        k_gather<<<blocks, 256, 0, stream>>>(feat, map, out);
    } else {
        int n4 = out_size / 4;
        k_zero<<<(n4 + 255) / 256, 256, 0, stream>>>(out, n4);
        int PC = P * CBEV;
        k_scatter_direct<<<(PC + 255) / 256, 256, 0, stream>>>(feat, coords, out, PC);
    }
}